// NeRFRenderer_43722767073673
// MI455X (gfx1250) — compile-verified
//
#include <hip/hip_runtime.h>

// NeRF importance sampling (inverse-CDF) for MI455X / gfx1250.
// R = 262144 rays, 127 weights/ray, 128 output samples/ray.
// Memory-bound (~274 MB @ 23.3 TB/s ~= 12 us). Single fused pass:
//   Phase 0: TDM tensor_load_to_lds DMAs each wave's contiguous 16x127 f32
//            weight tile (8128 B) into LDS (TENSORcnt), overlapped with the
//            per-ray near/far computation (global loads + divides).
//   Phase A: per-wave prefix scan of 16 rays via V_WMMA_F32_16X16X4_F32
//            (S_c = (W_c+eps) x UpperTriOnes, 8 chunks, 4 WMMAs each), in-place.
//   Phase B: lanes 0-15 combine chunk totals -> normalized offsets; all lanes
//            materialize the normalized CDF in LDS in place.
//   Phase C: 7-step binary search per sample over the LDS CDF, coalesced stores.

typedef __attribute__((ext_vector_type(2))) float v2f;
typedef __attribute__((ext_vector_type(8))) float v8f;
typedef __attribute__((ext_vector_type(4))) unsigned int u32x4;
typedef __attribute__((ext_vector_type(8))) int i32x8;
typedef __attribute__((ext_vector_type(4))) int i32x4;

#define W_PER_RAY 127
#define N_SAMP 128
#define RAYS_PER_WAVE 16
#define WAVES_PER_BLOCK 4
#define RAYS_PER_BLOCK (RAYS_PER_WAVE * WAVES_PER_BLOCK)
#define TILE_WORDS (RAYS_PER_WAVE * W_PER_RAY) /* 2032 f32 = 8128 B, contiguous */

#if __has_builtin(__builtin_amdgcn_tensor_load_to_lds)
#define HAVE_TDM 1
#endif

__global__ __launch_bounds__(WAVES_PER_BLOCK * 32)
void nerf_sample_pdf_wmma(const float* __restrict__ rays_o,
                          const float* __restrict__ rays_d,
                          const float* __restrict__ weights,
                          const int* __restrict__ bound_p,
                          float* __restrict__ out,
                          int R)
{
    // Per-wave LDS regions (packed row stride 127: odd stride => conflict-free
    // column access for WMMA A-fragment gathers).
    __shared__ __align__(16) float sW[WAVES_PER_BLOCK][TILE_WORDS]; // weights -> scans -> cdf
    __shared__ float sOffn[WAVES_PER_BLOCK][RAYS_PER_WAVE * 8];     // normalized chunk offsets
    __shared__ float sNear[WAVES_PER_BLOCK][RAYS_PER_WAVE];
    __shared__ float sScale[WAVES_PER_BLOCK][RAYS_PER_WAVE];
    __shared__ float sInv[WAVES_PER_BLOCK][RAYS_PER_WAVE];

    const int lane = threadIdx.x & 31;
    const int wv   = threadIdx.x >> 5;
    const int r0   = (blockIdx.x * WAVES_PER_BLOCK + wv) * RAYS_PER_WAVE;
    if (r0 >= R) return;

    float* W = sW[wv];

    // ---------------- Phase 0: weight tile -> LDS ----------------------------
#ifdef HAVE_TDM
    {
        // D# for a 1D copy: 2032 f32 from (weights + r0*127) to this wave's
        // LDS tile. Bit layout per CDNA5 ISA ch.8 (group0 / group1).
        const unsigned long long ga =
            (unsigned long long)(const void*)(weights + (size_t)r0 * W_PER_RAY);
        const unsigned int lds_addr = (unsigned int)(unsigned long long)(uintptr_t)W;

        u32x4 g0;
        g0.x = 1u;                                   // count=1, no gather, user D#
        g0.y = lds_addr;                             // [63:32] lds_addr (bytes)
        g0.z = (unsigned int)(ga & 0xffffffffull);   // [95:64] global_addr lo
        g0.w = (unsigned int)((ga >> 32) & 0x01ffffffull) // [120:96] global_addr hi
               | 0x80000000u;                        // [127:126] type=2 (image)

        i32x8 g1;
        g1.s0 = (int)(2u << 16);                     // [17:16] data_size=2 (4 B)
        g1.s1 = (int)((TILE_WORDS & 0xffffu) << 16); // [63:48] tensor_dim0 lo16
        g1.s2 = (int)(1u << 16);                     // [79:64] dim0 hi16=0; [95:80] tensor_dim1=1
        g1.s3 = (int)((TILE_WORDS & 0xffffu) << 16); // [111:96] dim1 hi=0; [127:112] tile_dim0
        g1.s4 = 1;                                   // [143:128] tile_dim1=1; [159:144] tile_dim2=0
        g1.s5 = (int)TILE_WORDS;                     // [191:160] tensor_dim0_stride lo32
        g1.s6 = 0;                                   // stride hi16; tensor_dim1_stride lo16
        g1.s7 = 0;

        i32x4 gz4 = {0, 0, 0, 0};                    // groups 2/3 unused (<=2D)
        i32x8 gz8 = {0, 0, 0, 0, 0, 0, 0, 0};        // extra group (clang-23 form)
        __builtin_amdgcn_tensor_load_to_lds(g0, g1, gz4, gz4, gz8, 0);
    }
#else
    {
        const float* g = weights + (size_t)r0 * W_PER_RAY;
        for (int i = lane; i < TILE_WORDS; i += 32)
            W[i] = g[i];
    }
#endif

    // ------- Overlap with DMA: per-ray near/far (global loads + divides) -----
    if (lane < 16) {
        const int   r   = lane;
        const int   ray = r0 + r;
        const float bnd = (float)(*bound_p);
        float nearv = -1e30f, farv = 1e30f;
#pragma unroll
        for (int ax = 0; ax < 3; ++ax) {
            const float o  = rays_o[(size_t)ray * 3 + ax];
            const float d  = rays_d[(size_t)ray * 3 + ax] + 1e-15f;
            const float t0 = (-bnd - o) / d;
            const float t1 = ( bnd - o) / d;
            nearv = fmaxf(nearv, fminf(t0, t1));
            farv  = fminf(farv,  fmaxf(t0, t1));
        }
        nearv = fmaxf(nearv, 0.05f);
        sNear[wv][r]  = nearv;
        sScale[wv][r] = (farv - nearv) * (1.0f / 127.0f); // bin step
    }

#ifdef HAVE_TDM
    __builtin_amdgcn_s_wait_tensorcnt(0);            // tile resident in LDS
#endif

    // ---------------- Phase A: chunk-local inclusive scans via WMMA ----------
    // S_c(16x16) = (W_c + EPS_W)(16x16) * U16 (upper-tri ones),
    // as 4 x (16x4)*(4x16) WMMA accumulations. EPS_W applied at fragment read
    // (each weight is gathered exactly once).
    const int rA   = lane & 15;      // A-matrix row M / B,D column N
    const int hi   = lane >> 4;      // lane half
    const int off0 = hi * 2;         // K offset inside a K=4 slice
    for (int c = 0; c < 8; ++c) {
        const int colbase = c * 16;
        v8f acc = {0.f, 0.f, 0.f, 0.f, 0.f, 0.f, 0.f, 0.f};
#pragma unroll
        for (int k = 0; k < 4; ++k) {
            const int col0 = colbase + 4 * k + off0;
            v2f a, b;
            a.x = (col0     < W_PER_RAY) ? (W[rA * W_PER_RAY + col0]     + 1e-5f) : 0.0f;
            a.y = (col0 + 1 < W_PER_RAY) ? (W[rA * W_PER_RAY + col0 + 1] + 1e-5f) : 0.0f;
            // U16[k][n] = (k <= n)
            b.x = (4 * k + off0     <= rA) ? 1.0f : 0.0f;
            b.y = (4 * k + off0 + 1 <= rA) ? 1.0f : 0.0f;
            acc = __builtin_amdgcn_wmma_f32_16x16x4_f32(
                false, a, false, b, (short)0, acc, false, false);
        }
        // Write S_c back in place. D VGPR v: row = v + 8*hi, col N = lane&15.
        // Skip col 127 (would alias next row; its value equals col 126).
#pragma unroll
        for (int v = 0; v < 8; ++v) {
            const int row = v + hi * 8;
            const int col = colbase + rA;
            if (col < W_PER_RAY)
                W[row * W_PER_RAY + col] = acc[v];
        }
    }

    // ---------------- Phase B1: per-ray chunk offsets + total ----------------
    if (lane < 16) {
        const int r = lane;
        const float* Wr = W + r * W_PER_RAY;
        float tot[8];
#pragma unroll
        for (int c = 0; c < 7; ++c) tot[c] = Wr[c * 16 + 15];
        tot[7] = Wr[126];                       // == S_7[15] since pad weight = 0
        float off = 0.0f, offc[8];
#pragma unroll
        for (int c = 0; c < 8; ++c) { offc[c] = off; off += tot[c]; }
        const float inv = 1.0f / off;           // off == sum(w + EPS_W)
        sInv[wv][r] = inv;
#pragma unroll
        for (int c = 0; c < 8; ++c) sOffn[wv][r * 8 + c] = offc[c] * inv;
    }

    // ---------------- Phase B2: materialize normalized CDF in place ----------
    // cdf1[j] := cdf[j+1] = (S[j] + chunk_off) * inv, j = 0..126.
    {
        const int r   = lane & 15;
        const int sub = lane >> 4;
        const float inv = sInv[wv][r];
        float* Wr = W + r * W_PER_RAY;
#pragma unroll
        for (int c = 0; c < 8; ++c) {
            const float oc = sOffn[wv][r * 8 + c];
#pragma unroll
            for (int j = 0; j < 8; ++j) {
                const int col = c * 16 + sub + 2 * j;
                if (col < W_PER_RAY)
                    Wr[col] = Wr[col] * inv + oc;
            }
        }
    }

    // ---------------- Phase C: inverse-CDF sampling (binary search) ----------
    // searchsorted(cdf, u, 'right'), cdf[0]=0 so start lo=1. Lanes map to
    // consecutive sample indices k -> fully coalesced output stores.
    const float u_step = 1.0f / (float)N_SAMP;
    for (int r = 0; r < RAYS_PER_WAVE; ++r) {
        const float nearv = sNear[wv][r];
        const float scale = sScale[wv][r];
        const float* cdf1 = W + r * W_PER_RAY;
        float* orow = out + (size_t)(r0 + r) * N_SAMP;
#pragma unroll
        for (int jj = 0; jj < N_SAMP / 32; ++jj) {
            const int   k = lane + 32 * jj;
            const float u = ((float)k + 0.5f) * u_step;
            int lo = 1, hiq = 128;
#pragma unroll
            for (int it = 0; it < 7; ++it) {
                const int   mid = (lo + hiq) >> 1;
                const float cv  = cdf1[mid - 1];     // cdf[mid]
                if (cv <= u) lo = mid + 1; else hiq = mid;
            }
            const int   inds   = lo;
            const int   below  = inds - 1;
            const int   above  = (inds < 127) ? inds : 127;
            const float cdf_lo = (below == 0) ? 0.0f : cdf1[below - 1];
            const float cdf_hi = cdf1[above - 1];
            float denom = cdf_hi - cdf_lo;
            denom = (denom < 1e-5f) ? 1.0f : denom;
            const float t = (u - cdf_lo) / denom;
            orow[k] = nearv + scale * ((float)below + t * (float)(above - below));
        }
    }
}

extern "C" void kernel_launch(void* const* d_in, const int* in_sizes, int n_in,
                              void* d_out, int out_size, void* d_ws, size_t ws_size,
                              hipStream_t stream) {
    const float* rays_o  = (const float*)d_in[0];
    const float* rays_d  = (const float*)d_in[1];
    const float* weights = (const float*)d_in[2];
    const int*   bound   = (const int*)d_in[3];
    // d_in[4] = n_samples (== 128, baked into the kernel structure)
    float* out = (float*)d_out;

    const int R = in_sizes[0] / 3;
    const int blocks = (R + RAYS_PER_BLOCK - 1) / RAYS_PER_BLOCK;
    nerf_sample_pdf_wmma<<<blocks, WAVES_PER_BLOCK * 32, 0, stream>>>(
        rays_o, rays_d, weights, bound, out, R);
}